// LinearEmulator_7370163880246
// MI455X (gfx1250) — compile-verified
//
#include <hip/hip_runtime.h>
#include <hip/hip_bf16.h>

// Pseudo-Voigt transmission product:
//   out[p] = poly(wl[p]) * prod_l (1 - V[l,p])
// Compute-bound (3e8 pair evals, 2 transcendentals each). Strategy:
//   1) precompute per-line folded constants (lam, A_lor, gamma^2, A_gau, k)
//   2) main kernel: 2 pixels/thread, lines staged in 512-line LDS tiles via
//      gfx1250 async global->LDS DMA, double-buffered on ASYNCcnt.
// Inner loop per (pixel,line): 7 VALU + 2 TRANS (v_rcp_f32, v_exp_f32),
// product folded as prod = fma(-v, prod, prod).

#define TILE  512         // lines per LDS tile
#define BLOCK 128         // threads per block (4 waves)
#define PIX_PER_BLOCK (2 * BLOCK)

static_assert(TILE == 4 * BLOCK, "copy pattern assumes TILE == 4*BLOCK");

#define LOG2E 1.4426950408889634f

// ---------------------------------------------------------------------------
// Per-line constant folding (runs once over ~3k lines; cost negligible).
// ---------------------------------------------------------------------------
__global__ void voigt_precompute(const float* __restrict__ lam,
                                 const float* __restrict__ amp,
                                 const float* __restrict__ sw,
                                 const float* __restrict__ gw,
                                 float4* __restrict__ p0,
                                 float*  __restrict__ p1,
                                 int nl, int nlpad) {
    int i = blockIdx.x * blockDim.x + threadIdx.x;
    if (i >= nlpad) return;
    if (i >= nl) {
        // neutral line: factor == 1.0 exactly (A_lor=A_gau=0, gamma2=1, k=0)
        p0[i] = make_float4(0.0f, 0.0f, 1.0f, 0.0f);
        p1[i] = 0.0f;
        return;
    }
    float sigma = __builtin_amdgcn_exp2f(sw[i] * LOG2E);
    float gamma = __builtin_amdgcn_exp2f(gw[i] * LOG2E);
    float A     = __builtin_amdgcn_exp2f(amp[i] * LOG2E);

    float fG = 2.3548f * sigma;
    float fL = 2.0f * gamma;
    float fG2 = fG * fG, fG3 = fG2 * fG, fG4 = fG3 * fG, fG5 = fG4 * fG;
    float fL2 = fL * fL, fL3 = fL2 * fL, fL4 = fL3 * fL, fL5 = fL4 * fL;
    float S = fG5
            + 2.69269f * fG4 * fL
            + 2.42843f * fG3 * fL2
            + 4.47163f * fG2 * fL3
            + 0.07842f * fG  * fL4
            + fL5;
    // S^(1/5) = exp2(log2(S)/5)
    float fwhm = __builtin_amdgcn_exp2f(0.2f * __builtin_amdgcn_logf(S));
    float f = fL / fwhm;
    float eta = ((0.11116f * f - 0.47719f) * f + 1.36603f) * f;

    float A_lor = A * eta * gamma * (1.0f / 3.141592654f);
    float A_gau = A * (1.0f - eta) / (sigma * 2.5066f);
    float k     = -0.5f * LOG2E / (sigma * sigma);   // exp(-0.5 t^2) = exp2(k*d^2)

    p0[i] = make_float4(lam[i], A_lor, gamma * gamma, A_gau);
    p1[i] = k;
}

// ---------------------------------------------------------------------------
// Async global->LDS tile copy (gfx1250 ASYNCcnt path).
// Generic shared pointers: low 32 bits == workgroup-relative LDS byte offset
// (LDS aperture, ISA 10.2), which is exactly what VDST wants.
// 5 async instructions per call -> ASYNCcnt += 5 per wave.
// ---------------------------------------------------------------------------
__device__ __forceinline__ void async_b128(const void* gptr, const void* sptr) {
    unsigned int       l = (unsigned int)(unsigned long long)sptr;
    unsigned long long a = (unsigned long long)gptr;
    asm volatile("global_load_async_to_lds_b128 %0, %1, off"
                 :: "v"(l), "v"(a) : "memory");
}

__device__ __forceinline__ void async_tile_copy(const float4* __restrict__ g0,
                                                const float*  __restrict__ g1,
                                                float4* s0, float* s1, int t) {
#pragma unroll
    for (int q = 0; q < 4; ++q)
        async_b128(g0 + t + q * BLOCK, s0 + t + q * BLOCK);
    async_b128(g1 + 4 * t, s1 + 4 * t);
}

// one (pixel,line) evaluation: 7 VALU + 2 TRANS
__device__ __forceinline__ void voigt_step(float w, float4 P, float k, float& prod) {
    float d   = w - P.x;
    float d2  = d * d;
    float lor = P.y * __builtin_amdgcn_rcpf(d2 + P.z);
    float gau = __builtin_amdgcn_exp2f(k * d2);
    float v   = fmaf(P.w, gau, lor);
    prod = fmaf(-v, prod, prod);          // prod *= (1 - v)
}

// ---------------------------------------------------------------------------
// Main kernel: two pixels per thread, product over all (padded) lines.
// ---------------------------------------------------------------------------
__global__ void __launch_bounds__(BLOCK)
voigt_prod(const float* __restrict__ wl,
           const float4* __restrict__ p0,
           const float*  __restrict__ p1,
           float* __restrict__ out,
           const float* __restrict__ pa,
           const float* __restrict__ pb,
           const float* __restrict__ pc,
           int npix, int nt) {
    __shared__ __align__(16) float4 sh0[2][TILE];
    __shared__ __align__(16) float  sh1[2][TILE];

    const int t   = threadIdx.x;
    const int ga  = blockIdx.x * PIX_PER_BLOCK + t;          // pixel A
    const int gb  = ga + BLOCK;                              // pixel B
    const float wa = (ga < npix) ? wl[ga] : 10000.0f;
    const float wb = (gb < npix) ? wl[gb] : 10000.0f;

    float prodA = 1.0f;
    float prodB = 1.0f;

    // prefetch tile 0 into buffer 0  (ASYNCcnt = 5)
    async_tile_copy(p0, p1, sh0[0], sh1[0], t);

    int cur = 0;
    for (int tile = 0; tile < nt; ++tile) {
        const int nxt = cur ^ 1;
        if (tile + 1 < nt) {
            // prefetch next tile into the other buffer (ASYNCcnt <= 10),
            // then wait until only the prefetch remains outstanding.
            async_tile_copy(p0 + (tile + 1) * TILE, p1 + (tile + 1) * TILE,
                            sh0[nxt], sh1[nxt], t);
            asm volatile("s_wait_asynccnt 5" ::: "memory");
        } else {
            asm volatile("s_wait_asynccnt 0" ::: "memory");
        }
        __syncthreads();   // current tile fully resident in LDS for all waves

        const float4* __restrict__ P0 = sh0[cur];
        const float4* __restrict__ K4 = (const float4*)sh1[cur];
#pragma unroll 2
        for (int j4 = 0; j4 < TILE / 4; ++j4) {
            float4 kq = K4[j4];                 // one ds_load_b128 for 4 k's
            float4 Pq0 = P0[4 * j4 + 0];
            float4 Pq1 = P0[4 * j4 + 1];
            float4 Pq2 = P0[4 * j4 + 2];
            float4 Pq3 = P0[4 * j4 + 3];
            voigt_step(wa, Pq0, kq.x, prodA);  voigt_step(wb, Pq0, kq.x, prodB);
            voigt_step(wa, Pq1, kq.y, prodA);  voigt_step(wb, Pq1, kq.y, prodB);
            voigt_step(wa, Pq2, kq.z, prodA);  voigt_step(wb, Pq2, kq.z, prodB);
            voigt_step(wa, Pq3, kq.w, prodA);  voigt_step(wb, Pq3, kq.w, prodB);
        }

        __syncthreads();   // everyone done reading buf[cur] before it's rewritten
        cur = nxt;
    }

    const float A = pa[0], B = pb[0], C = pc[0];
    if (ga < npix) {
        float x = (wa - 10500.0f) * (1.0f / 2500.0f);
        out[ga] = prodA * fmaf(fmaf(C, x, B), x, A);
    }
    if (gb < npix) {
        float x = (wb - 10500.0f) * (1.0f / 2500.0f);
        out[gb] = prodB * fmaf(fmaf(C, x, B), x, A);
    }
}

// ---------------------------------------------------------------------------
// Host entry point. Inputs (setup_inputs order):
//   0: wl (npix f32)  1: lam_centers (nl)  2: amplitudes (nl)
//   3: sigma_widths (nl)  4: gamma_widths (nl)
//   5: a_coeff (1)  6: b_coeff (1)  7: c_coeff (1)
// ---------------------------------------------------------------------------
extern "C" void kernel_launch(void* const* d_in, const int* in_sizes, int n_in,
                              void* d_out, int out_size, void* d_ws, size_t ws_size,
                              hipStream_t stream) {
    const float* wl  = (const float*)d_in[0];
    const float* lam = (const float*)d_in[1];
    const float* amp = (const float*)d_in[2];
    const float* sw  = (const float*)d_in[3];
    const float* gw  = (const float*)d_in[4];
    const float* pa  = (const float*)d_in[5];
    const float* pb  = (const float*)d_in[6];
    const float* pc  = (const float*)d_in[7];
    float* out = (float*)d_out;

    const int npix  = in_sizes[0];
    const int nl    = in_sizes[1];
    const int nt    = (nl + TILE - 1) / TILE;
    const int nlpad = nt * TILE;

    float4* p0 = (float4*)d_ws;
    float*  p1 = (float*)((char*)d_ws + (size_t)nlpad * sizeof(float4));

    voigt_precompute<<<(nlpad + 255) / 256, 256, 0, stream>>>(
        lam, amp, sw, gw, p0, p1, nl, nlpad);

    voigt_prod<<<(npix + PIX_PER_BLOCK - 1) / PIX_PER_BLOCK, BLOCK, 0, stream>>>(
        wl, p0, p1, out, pa, pb, pc, npix, nt);
}